// MultiHeadGeneticAttention_43413529428368
// MI455X (gfx1250) — compile-verified
//
#include <hip/hip_runtime.h>

typedef float v2f __attribute__((ext_vector_type(2)));
typedef float v4f __attribute__((ext_vector_type(4)));
typedef float v8f __attribute__((ext_vector_type(8)));
typedef unsigned int u32x4 __attribute__((ext_vector_type(4)));
typedef int i32x4 __attribute__((ext_vector_type(4)));
typedef int i32x8 __attribute__((ext_vector_type(8)));

#define B_   2
#define T_   2048
#define D_   1024
#define H_   16
#define HD_  64
#define WIN_ 512

#if defined(__has_builtin)
#  if __has_builtin(__builtin_amdgcn_tensor_load_to_lds)
#    define HAVE_TDM 1
#  endif
#endif
#ifndef HAVE_TDM
#  define HAVE_TDM 0
#endif

// ---------------------------------------------------------------------------
// TDM 2D tile load: global (row-major, elem stride0 between rows) -> LDS with
// per-row DWORD padding.  D# built per CDNA5 ISA 8.3/8.4:
//   group0: [1:0]=count=1 | lds_addr | global_addr[56:0] | type=2
//   group1: data_size=2(4B), pad_enable, pad_interval (2^(p+1) DW),
//           pad_amount (n+1 DW), tensor dims == tile dims (no OOB),
//           tensor_dim0_stride = row stride in elements.
// ---------------------------------------------------------------------------
#if HAVE_TDM
__device__ __forceinline__ void tdm_load_2d(unsigned lds_addr, const void* gptr,
                                            unsigned tile_d0, unsigned tile_d1,
                                            unsigned long long stride0_elems,
                                            unsigned pad_interval_code,
                                            unsigned pad_amount_code)
{
    unsigned long long ga = (unsigned long long)gptr;
    u32x4 g0;
    g0.x = 1u;                                            // count=1, user D#
    g0.y = lds_addr;                                      // LDS byte address
    g0.z = (unsigned)(ga & 0xFFFFFFFFu);                  // global_addr[31:0]
    g0.w = (unsigned)((ga >> 32) & 0x01FFFFFFu) | (2u << 30);  // [56:32]|type=2

    i32x8 g1;
    g1[0] = (int)((2u << 16) | (1u << 20) |               // data_size=4B, pad_en
                  (pad_interval_code << 22) | (pad_amount_code << 25));
    g1[1] = (int)((tile_d0 & 0xFFFFu) << 16);             // tensor_dim0[15:0]
    g1[2] = (int)((tile_d0 >> 16) | ((tile_d1 & 0xFFFFu) << 16));
    g1[3] = (int)((tile_d1 >> 16) | (tile_d0 << 16));     // tile_dim0
    g1[4] = (int)(tile_d1);                               // tile_dim1, tile_dim2=0
    g1[5] = (int)(stride0_elems & 0xFFFFFFFFu);           // dim0_stride[31:0]
    g1[6] = (int)((stride0_elems >> 32) & 0xFFFFu);       // [47:32], dim1_stride=0
    g1[7] = 0;

    i32x4 z4 = {};
#if __clang_major__ >= 23
    i32x8 z8 = {};
    __builtin_amdgcn_tensor_load_to_lds(g0, g1, z4, z4, z8, 0);
#else
    __builtin_amdgcn_tensor_load_to_lds(g0, g1, z4, z4, 0);
#endif
}
#endif

// ---------------------------------------------------------------------------
// GEMM: C(M,N) = A(M,K) @ W(K,N) + bias, fp32 via V_WMMA_F32_16X16X4_F32.
// Block tile 64x64, K-step 32, 256 threads = 8 waves (2x4); each wave owns a
// 32x16 subtile (two accumulators).  Double-buffered LDS tiles filled by the
// Tensor Data Mover (wave 0 issues next tile while all waves compute), synced
// with s_wait_tensorcnt + barrier.  +4-float row padding (conflict-free
// fragment reads) reproduced by the TDM pad fields.
// scatter_bhtd=1 writes C in (B,H,T,HD) layout for g/v.
// ---------------------------------------------------------------------------
#define TM 64
#define TN 64
#define TK 32
#define PAD 4

__global__ __launch_bounds__(256)
void gemm_f32_wmma(const float* __restrict__ A, const float* __restrict__ W,
                   const float* __restrict__ bias, float* __restrict__ C,
                   int M, int N, int K, int scatter_bhtd)
{
    __shared__ float As[2][TM][TK + PAD];   // 2 x 64 x 36
    __shared__ float Bs[2][TK][TN + PAD];   // 2 x 32 x 68

    const int tid   = threadIdx.x;
    const int lane  = tid & 31;
    const int wave  = tid >> 5;
    const int sub_m = (wave >> 2) * 32;   // 0,32
    const int sub_n = (wave & 3)  * 16;   // 0..48

    const int m0 = blockIdx.y * TM;
    const int n0 = blockIdx.x * TN;

    const int lm = lane & 15;             // M/N index within fragment
    const int lk = (lane >> 4) * 2;       // K sub-offset (0 or 2)

    const float* Abase = A + (size_t)m0 * K;        // + k0
    const float* Wbase = W + n0;                    // + k0*N

#if HAVE_TDM
    // --- prologue: TDM-fill buffer 0 with the k0=0 tiles ---
    if (wave == 0) {
        tdm_load_2d((unsigned)(size_t)&As[0][0][0], Abase,
                    TK, TM, (unsigned long long)K, 4u, 3u);   // 32 DW rows +4
        tdm_load_2d((unsigned)(size_t)&Bs[0][0][0], Wbase,
                    TN, TK, (unsigned long long)N, 5u, 3u);   // 64 DW rows +4
    }
    __builtin_amdgcn_s_wait_tensorcnt(0);
    __syncthreads();
#else
    const int arow = tid >> 3, acol = (tid & 7) * 4;
    const int brow = tid >> 4, bcol = (tid & 15) * 4;
    const float* ag0 = A + (size_t)(m0 + arow) * K + acol;
    const float* ag1 = ag0 + (size_t)32 * K;
    const float* wg0 = W + (size_t)brow * N + n0 + bcol;
    const float* wg1 = wg0 + (size_t)16 * N;
#endif

    v8f acc0 = {}, acc1 = {};
    int buf = 0;

    for (int k0 = 0; k0 < K; k0 += TK) {
#if HAVE_TDM
        if ((k0 + TK < K) && wave == 0) {            // async-fill next buffer
            tdm_load_2d((unsigned)(size_t)&As[buf ^ 1][0][0],
                        Abase + (k0 + TK),
                        TK, TM, (unsigned long long)K, 4u, 3u);
            tdm_load_2d((unsigned)(size_t)&Bs[buf ^ 1][0][0],
                        Wbase + (size_t)(k0 + TK) * N,
                        TN, TK, (unsigned long long)N, 5u, 3u);
        }
#else
        *(v4f*)&As[buf][arow][acol]      = *(const v4f*)(ag0 + k0);
        *(v4f*)&As[buf][arow + 32][acol] = *(const v4f*)(ag1 + k0);
        *(v4f*)&Bs[buf][brow][bcol]      = *(const v4f*)(wg0 + (size_t)k0 * N);
        *(v4f*)&Bs[buf][brow + 16][bcol] = *(const v4f*)(wg1 + (size_t)k0 * N);
        __syncthreads();
#endif

        #pragma unroll
        for (int kb = 0; kb < TK; kb += 4) {
            v2f a0, a1, b;
            a0.x = As[buf][sub_m + lm][kb + lk + 0];
            a0.y = As[buf][sub_m + lm][kb + lk + 1];
            a1.x = As[buf][sub_m + 16 + lm][kb + lk + 0];
            a1.y = As[buf][sub_m + 16 + lm][kb + lk + 1];
            b.x  = Bs[buf][kb + lk + 0][sub_n + lm];
            b.y  = Bs[buf][kb + lk + 1][sub_n + lm];
            acc0 = __builtin_amdgcn_wmma_f32_16x16x4_f32(
                       false, a0, false, b, (short)0, acc0, false, false);
            acc1 = __builtin_amdgcn_wmma_f32_16x16x4_f32(
                       false, a1, false, b, (short)0, acc1, false, false);
        }

#if HAVE_TDM
        __builtin_amdgcn_s_wait_tensorcnt(0);        // next tile landed (wave 0)
        __syncthreads();                             // publish; readers done
        buf ^= 1;
#else
        __syncthreads();
#endif
    }

    const int   n  = n0 + sub_n + lm;
    const float bn = bias[n];
    #pragma unroll
    for (int mm = 0; mm < 2; ++mm) {
        #pragma unroll
        for (int r = 0; r < 8; ++r) {
            int   m   = m0 + sub_m + mm * 16 + r + (lane >> 4) * 8;
            float val = (mm ? acc1[r] : acc0[r]) + bn;
            if (scatter_bhtd) {            // (m,n) -> (B,H,T,HD)
                int bb = m / T_, t = m % T_;
                int h  = n >> 6, d = n & 63;
                C[(((size_t)bb * H_ + h) * T_ + t) * HD_ + d] = val;
            } else {
                C[(size_t)m * N + n] = val;
            }
        }
    }
}

// ---------------------------------------------------------------------------
// Sliding-window stats: running window sum of g along T, then
// fit = (1/denom)/sum_inv and rowinv = 1/max(fit . wsum, 1e-10).
// One block per (b,h); 64 threads own the 64 head dims.
// ---------------------------------------------------------------------------
__global__ __launch_bounds__(64)
void window_fit(const float* __restrict__ g, float* __restrict__ fit,
                float* __restrict__ rowinv)
{
    const int    bh   = blockIdx.x;
    const int    d    = threadIdx.x;   // 0..63
    const int    lane = d & 31;
    const int    wv   = d >> 5;
    const size_t base = (size_t)bh * T_ * HD_;

    __shared__ float red[2];
    __shared__ float red2[2];

    float wsum = 0.0f;
    for (int i = 0; i < T_; ++i) {
        wsum += g[base + (size_t)i * HD_ + d];
        if (i >= WIN_) wsum -= g[base + (size_t)(i - WIN_) * HD_ + d];
        float cnt   = (float)((i + 1) < WIN_ ? (i + 1) : WIN_);
        float denom = wsum / cnt + 0.5f;
        float inv   = 1.0f / denom;

        float s = inv;                                   // reduce inv over 64
        #pragma unroll
        for (int off = 16; off > 0; off >>= 1) s += __shfl_xor(s, off, 32);
        if (lane == 0) red[wv] = s;
        __syncthreads();
        float sum_inv = red[0] + red[1];

        float f  = inv / sum_inv;
        float rs = f * wsum;                             // reduce fit.wsum
        #pragma unroll
        for (int off = 16; off > 0; off >>= 1) rs += __shfl_xor(rs, off, 32);
        if (lane == 0) red2[wv] = rs;
        __syncthreads();
        float row = red2[0] + red2[1];
        row = row > 1e-10f ? row : 1e-10f;

        fit[base + (size_t)i * HD_ + d] = f;
        if (d == 0) rowinv[(size_t)bh * T_ + i] = 1.0f / row;
        __syncthreads();                                 // LDS reuse next iter
    }
}

// ---------------------------------------------------------------------------
// Banded attention: one wave per 16-row query tile of one (b,h).
// Stage 1: S(16x16) = Fit_i(16x64) . G_j(16x64)^T   -> 16 f32 WMMAs
// mask band, spill S to LDS to re-fragment as an A operand,
// Stage 2: O(16x64) += S(16x16) . V_j(16x64)        -> 16 f32 WMMAs
// Finally scale rows by rowinv and write (B,T,D) layout for the Wo GEMM.
// ---------------------------------------------------------------------------
__global__ __launch_bounds__(32)
void banded_attn(const float* __restrict__ fit, const float* __restrict__ g,
                 const float* __restrict__ v,   const float* __restrict__ rowinv,
                 float* __restrict__ out_pre)
{
    const int    i0   = blockIdx.x * 16;
    const int    bh   = blockIdx.y;
    const int    bb   = bh / H_;
    const int    h    = bh % H_;
    const size_t base = (size_t)bh * T_ * HD_;

    const int lane = threadIdx.x;
    const int lm   = lane & 15;
    const int lk   = (lane >> 4) * 2;

    __shared__ float sT[16][17];

    v2f fa[16];                                          // Fit A-fragments
    const float* frow = fit + base + (size_t)(i0 + lm) * HD_ + lk;
    #pragma unroll
    for (int ks = 0; ks < 16; ++ks)
        fa[ks] = *(const v2f*)(frow + ks * 4);           // 8B aligned (lk even)

    v8f o[4] = {};

    int lo = i0 - WIN_ + 1; if (lo < 0) lo = 0;
    const int jt_lo = lo >> 4;
    const int jt_hi = i0 >> 4;

    for (int jt = jt_lo; jt <= jt_hi; ++jt) {
        const int j0 = jt * 16;

        v8f s = {};                                      // S = Fit . G^T
        const float* grow = g + base + (size_t)(j0 + lm) * HD_ + lk;
        #pragma unroll
        for (int ks = 0; ks < 16; ++ks) {
            v2f gb = *(const v2f*)(grow + ks * 4);
            s = __builtin_amdgcn_wmma_f32_16x16x4_f32(
                    false, fa[ks], false, gb, (short)0, s, false, false);
        }

        #pragma unroll
        for (int r = 0; r < 8; ++r) {                    // band mask -> LDS
            int   mi = r + (lane >> 4) * 8;
            int   i  = i0 + mi;
            int   j  = j0 + lm;
            float sv = ((j <= i) && (j > i - WIN_)) ? s[r] : 0.0f;
            sT[mi][lm] = sv;
        }

        #pragma unroll
        for (int ks = 0; ks < 4; ++ks) {                 // O += S . V
            v2f a2;
            a2.x = sT[lm][ks * 4 + lk + 0];
            a2.y = sT[lm][ks * 4 + lk + 1];
            const float* vrow = v + base + (size_t)(j0 + ks * 4 + lk) * HD_ + lm;
            #pragma unroll
            for (int c = 0; c < 4; ++c) {
                v2f b2;
                b2.x = vrow[c * 16];
                b2.y = vrow[HD_ + c * 16];
                o[c] = __builtin_amdgcn_wmma_f32_16x16x4_f32(
                           false, a2, false, b2, (short)0, o[c], false, false);
            }
        }
    }

    #pragma unroll
    for (int r = 0; r < 8; ++r) {
        int   i    = i0 + r + (lane >> 4) * 8;
        float rinv = rowinv[(size_t)bh * T_ + i];
        #pragma unroll
        for (int c = 0; c < 4; ++c) {
            out_pre[((size_t)bb * T_ + i) * D_ + h * HD_ + c * 16 + lm]
                = o[c][r] * rinv;
        }
    }
}

// ---------------------------------------------------------------------------
extern "C" void kernel_launch(void* const* d_in, const int* in_sizes, int n_in,
                              void* d_out, int out_size, void* d_ws, size_t ws_size,
                              hipStream_t stream)
{
    const float* x  = (const float*)d_in[0];
    const float* Wg = (const float*)d_in[1];
    const float* bg = (const float*)d_in[2];
    const float* Wv = (const float*)d_in[3];
    const float* bv = (const float*)d_in[4];
    const float* Wo = (const float*)d_in[5];
    const float* bo = (const float*)d_in[6];
    float* out = (float*)d_out;

    const size_t MB = 1024u * 1024u;
    char*  ws   = (char*)d_ws;
    float* g    = (float*)(ws);                 // (B,H,T,64)  16 MB
    float* v    = (float*)(ws + 16 * MB);       // (B,H,T,64)  16 MB
    float* fit  = (float*)(ws + 32 * MB);       // (B,H,T,64)  16 MB
    float* pre  = (float*)(ws + 48 * MB);       // (B,T,D)     16 MB
    float* rinv = (float*)(ws + 64 * MB);       // (B,H,T)    256 KB

    dim3 gg(D_ / TN, (B_ * T_) / TM);           // (16,64)
    gemm_f32_wmma<<<gg, 256, 0, stream>>>(x, Wg, bg, g, B_ * T_, D_, D_, 1);
    gemm_f32_wmma<<<gg, 256, 0, stream>>>(x, Wv, bv, v, B_ * T_, D_, D_, 1);

    window_fit<<<B_ * H_, 64, 0, stream>>>(g, fit, rinv);

    dim3 ga(T_ / 16, B_ * H_);                  // (128,32)
    banded_attn<<<ga, 32, 0, stream>>>(fit, g, v, rinv, pre);

    gemm_f32_wmma<<<gg, 256, 0, stream>>>(pre, Wo, bo, out, B_ * T_, D_, D_, 0);
}